// LGCN_Encoder_12429635354867
// MI455X (gfx1250) — compile-verified
//
#include <hip/hip_runtime.h>

#define EMB 64

typedef float v2f __attribute__((ext_vector_type(2)));
typedef float v8f __attribute__((ext_vector_type(8)));

// ---------------------------------------------------------------------------
// init: x = acc = concat(user_emb, item_emb)   (float4 vectorized)
// ---------------------------------------------------------------------------
__global__ void lgcn_init(const float* __restrict__ ue, const float* __restrict__ ie,
                          float* __restrict__ x, float* __restrict__ acc,
                          int nUserF4, int nTotalF4) {
  int i = blockIdx.x * blockDim.x + threadIdx.x;
  if (i >= nTotalF4) return;
  float4 v = (i < nUserF4) ? ((const float4*)ue)[i]
                           : ((const float4*)ie)[i - nUserF4];
  ((float4*)x)[i]   = v;
  ((float4*)acc)[i] = v;
}

// ---------------------------------------------------------------------------
// zero: y = 0
// ---------------------------------------------------------------------------
__global__ void lgcn_zero(float* __restrict__ y, int nTotalF4) {
  int i = blockIdx.x * blockDim.x + threadIdx.x;
  if (i >= nTotalF4) return;
  ((float4*)y)[i] = make_float4(0.f, 0.f, 0.f, 0.f);
}

// ---------------------------------------------------------------------------
// scatter: for each edge e: y[rows[e]] += vals[e] * x[cols[e]]
// One edge per wave32: lane l owns float2 at column 2l (64 floats per row).
// Edge metadata forced scalar via readfirstlane; hw fp32 atomics into L2.
// ---------------------------------------------------------------------------
__global__ void lgcn_scatter(const int* __restrict__ rows, const int* __restrict__ cols,
                             const float* __restrict__ vals, const float* __restrict__ x,
                             float* __restrict__ y, int nnz) {
  const int lane   = threadIdx.x & 31;
  const int wave   = (blockIdx.x * blockDim.x + threadIdx.x) >> 5;
  const int nwaves = (gridDim.x * blockDim.x) >> 5;

  for (int e = wave; e < nnz; e += nwaves) {
    const int eu = __builtin_amdgcn_readfirstlane(e);   // wave-uniform index
    if (eu + nwaves < nnz) {                            // gfx1250 prefetch path
      __builtin_prefetch(rows + eu + nwaves, 0, 0);
      __builtin_prefetch(cols + eu + nwaves, 0, 0);
      __builtin_prefetch(vals + eu + nwaves, 0, 0);
    }
    const int   r = rows[eu];
    const int   c = cols[eu];
    const float v = vals[eu];

    const float2 xv = *(const float2*)(x + (size_t)c * EMB + lane * 2);
    float* dst = y + (size_t)r * EMB + lane * 2;
    unsafeAtomicAdd(dst,     v * xv.x);   // global_atomic_add_f32
    unsafeAtomicAdd(dst + 1, v * xv.y);
  }
}

// ---------------------------------------------------------------------------
// acc += y   (float4 vectorized)
// ---------------------------------------------------------------------------
__global__ void lgcn_acc(float* __restrict__ acc, const float* __restrict__ y,
                         int nTotalF4) {
  int i = blockIdx.x * blockDim.x + threadIdx.x;
  if (i >= nTotalF4) return;
  float4 a = ((float4*)acc)[i];
  float4 b = ((const float4*)y)[i];
  a.x += b.x; a.y += b.y; a.z += b.z; a.w += b.w;
  ((float4*)acc)[i] = a;
}

// ---------------------------------------------------------------------------
// finalize: acc *= 0.25, exactly, via chained V_WMMA_F32_16X16X4_F32.
// Per 16x16 tile T: D = sum_k A_k * B_k, A_k = T[:, 4k:4k+4],
// B_k[j,n] = 0.25 iff n == 4k+j else 0  =>  D = 0.25 * T  (bit-exact:
// every term is 0*x or 0.25*x). One tile per wave; tile guard is
// wave-uniform so EXEC is all-ones for the WMMAs.
//
// Operand layouts per ISA 7.12.2 (32-bit, wave32):
//   A 16x4 : M = lane&15, K = vgpr + 2*(lane>>4)
//   B 4x16 : N = lane&15, K = vgpr + 2*(lane>>4)
//   C/D    : vgpr r -> (M = r + 8*(lane>>4), N = lane&15)
// ---------------------------------------------------------------------------
__global__ void lgcn_finalize(float* __restrict__ acc, int nRowTiles) {
  const int wavesPerBlock = blockDim.x >> 5;
  const int tile = blockIdx.x * wavesPerBlock + (threadIdx.x >> 5);
  const int nTiles = nRowTiles * (EMB / 16);
  if (tile >= nTiles) return;                 // uniform per wave

  const int row0 = (tile >> 2) << 4;          // (tile / 4) * 16
  const int col0 = (tile & 3) << 4;           // (tile % 4) * 16
  const int lane = threadIdx.x & 31;
  const int m    = lane & 15;
  const int hi   = lane >> 4;

  const float* src = acc + (size_t)(row0 + m) * EMB + col0;

  v8f c = {0.f, 0.f, 0.f, 0.f, 0.f, 0.f, 0.f, 0.f};
#pragma unroll
  for (int k = 0; k < 4; ++k) {
    v2f a, b;
    a[0] = src[4 * k + 2 * hi + 0];
    a[1] = src[4 * k + 2 * hi + 1];
    b[0] = (m == (4 * k + 2 * hi + 0)) ? 0.25f : 0.0f;
    b[1] = (m == (4 * k + 2 * hi + 1)) ? 0.25f : 0.0f;
    c = __builtin_amdgcn_wmma_f32_16x16x4_f32(false, a, false, b,
                                              (short)0, c, false, false);
  }

  float* dst = acc + (size_t)(row0 + 8 * hi) * EMB + col0 + m;
#pragma unroll
  for (int r = 0; r < 8; ++r) {
    dst[(size_t)r * EMB] = c[r];
  }
}

// ---------------------------------------------------------------------------
// Host launcher
// inputs: [0] user_emb f32, [1] item_emb f32, [2] adj_rows i32,
//         [3] adj_cols i32, [4] adj_vals f32
// output: [N, 64] f32 (users then items, contiguous = full matrix)
// ---------------------------------------------------------------------------
extern "C" void kernel_launch(void* const* d_in, const int* in_sizes, int n_in,
                              void* d_out, int out_size, void* d_ws, size_t ws_size,
                              hipStream_t stream) {
  const float* ue   = (const float*)d_in[0];
  const float* ie   = (const float*)d_in[1];
  const int*   rows = (const int*)d_in[2];
  const int*   cols = (const int*)d_in[3];
  const float* vals = (const float*)d_in[4];

  const int nUsers = in_sizes[0] / EMB;
  const int nItems = in_sizes[1] / EMB;
  const int N      = nUsers + nItems;          // 150000
  const int nnz    = in_sizes[2];              // 8,000,000

  const int nTotalF4 = N * (EMB / 4);          // 2,400,000
  const int nUserF4  = nUsers * (EMB / 4);

  float* acc = (float*)d_out;                  // running layer sum lives in d_out
  float* x   = (float*)d_ws;                   // current ego embeddings
  float* y   = x + (size_t)N * EMB;            // next-layer scratch

  const int TB = 256;
  const int ewBlocks = (nTotalF4 + TB - 1) / TB;

  lgcn_init<<<ewBlocks, TB, 0, stream>>>(ue, ie, x, acc, nUserF4, nTotalF4);

  for (int layer = 0; layer < 3; ++layer) {
    lgcn_zero<<<ewBlocks, TB, 0, stream>>>(y, nTotalF4);
    lgcn_scatter<<<8192, TB, 0, stream>>>(rows, cols, vals, x, y, nnz);
    lgcn_acc<<<ewBlocks, TB, 0, stream>>>(acc, y, nTotalF4);
    float* t = x; x = y; y = t;                // next layer reads previous output
  }

  const int nRowTiles = N / 16;                // 9375 (exact)
  const int nTiles    = nRowTiles * (EMB / 16);
  const int wavesPB   = TB / 32;
  const int fBlocks   = (nTiles + wavesPB - 1) / wavesPB;
  lgcn_finalize<<<fBlocks, TB, 0, stream>>>(acc, nRowTiles);
}